// PackedRFTracer_91328184582334
// MI455X (gfx1250) — compile-verified
//
#include <hip/hip_runtime.h>
#include <stdint.h>

// ---------------------------------------------------------------------------
// MI455X (gfx1250) packed ray-marcher.
// Bandwidth-bound (~235 MB @ 23.3 TB/s ≈ 10 us); no matrix math -> no WMMA.
// CDNA5 paths: GLOBAL_LOAD_ASYNC_TO_LDS_B32 double-buffered (ASYNCcnt),
// s_wait_asynccnt N as the pipeline fence, wave32 ds_bpermute scan/reduction.
// ---------------------------------------------------------------------------

#define BLOCK 256
#define WAVES_PER_BLOCK (BLOCK / 32)
#define CHUNK 32
#define FIELDS 6  // density, delta, depth, r, g, b

// ---- CDNA5 async global->LDS copy (per-lane b32), signature probe-verified
#if defined(__AMDGCN__) && __has_builtin(__builtin_amdgcn_global_load_async_to_lds_b32)
#define ASYNC_CP_B32(GSRC, LDST)                                               \
  __builtin_amdgcn_global_load_async_to_lds_b32(                               \
      (__attribute__((address_space(1))) int*)(GSRC),                          \
      (__attribute__((address_space(3))) int*)(LDST), 0, 0)
#else
__device__ __forceinline__ void __async_cp_b32(const void* g, void* l) {
  unsigned loff = (unsigned)(uintptr_t)(__attribute__((address_space(3))) void*)l;
  asm volatile("global_load_async_to_lds_b32 %0, %1, off"
               :: "v"(loff), "v"(g) : "memory");
}
#define ASYNC_CP_B32(GSRC, LDST) __async_cp_b32((const void*)(GSRC), (void*)(LDST))
#endif

#if defined(__AMDGCN__) && __has_builtin(__builtin_amdgcn_s_wait_asynccnt)
#define WAIT_ASYNC(N) __builtin_amdgcn_s_wait_asynccnt(N)
#else
#define WAIT_ASYNC(N) asm volatile("s_wait_asynccnt %0" :: "i"(N) : "memory")
#endif

// ---- wave32 lane exchange via ds_bpermute (no header warpSize dependence) -
__device__ __forceinline__ float bperm_f32(int srclane, float v) {
  int r = __builtin_amdgcn_ds_bpermute(srclane << 2, __builtin_bit_cast(int, v));
  return __builtin_bit_cast(float, r);
}

// ---------------------------------------------------------------------------
__global__ void init_seg_kernel(int* __restrict__ seg_start,
                                int* __restrict__ seg_end, int R) {
  int r = blockIdx.x * blockDim.x + threadIdx.x;
  if (r < R) { seg_start[r] = 0; seg_end[r] = 0; }
}

__global__ void mark_bounds_kernel(const int* __restrict__ ridx, int N,
                                   int* __restrict__ seg_start,
                                   int* __restrict__ seg_end) {
  int i = blockIdx.x * blockDim.x + threadIdx.x;
  if (i >= N) return;
  int r = ridx[i];
  if (i == 0 || ridx[i - 1] != r) seg_start[r] = i;            // pack boundary
  if (i == N - 1 || ridx[i + 1] != r) seg_end[r] = i + 1;      // pack end
}

// One wave per ray; 32-sample chunks; double-buffered async global->LDS
// staging so each wave keeps 2 chunks (1.5 KB) of HBM traffic in flight.
__global__ __launch_bounds__(BLOCK) void trace_kernel(
    const float* __restrict__ color,   // [N,3]
    const float* __restrict__ density, // [N]
    const float* __restrict__ deltas,  // [N]
    const float* __restrict__ depths,  // [N]
    const int* __restrict__ seg_start, // [R]
    const int* __restrict__ seg_end,   // [R]
    float* __restrict__ out,           // rgb[3R] | depth[R] | alpha[R] | hit[R]
    int R) {
  __shared__ float smem[WAVES_PER_BLOCK][2][FIELDS * CHUNK];  // SoA, no conflicts
  const int lane = threadIdx.x & 31;
  const int wave = threadIdx.x >> 5;
  const int ray = blockIdx.x * WAVES_PER_BLOCK + wave;
  if (ray >= R) return;

  const int s = seg_start[ray];
  const int e = seg_end[ray];
  const int nchunks = (e > s) ? ((e - s + CHUNK - 1) >> 5) : 0;

  // issue one chunk's 6 async b32 streams into buffer `sel`
  auto issue = [&](int base, int sel) {
    const int i = base + lane;
    const int j = (i < e) ? i : (e - 1);   // clamp: keep loads in-bounds
    float* b = &smem[wave][sel][0];
    ASYNC_CP_B32(density + j,       b + 0 * CHUNK + lane);
    ASYNC_CP_B32(deltas + j,        b + 1 * CHUNK + lane);
    ASYNC_CP_B32(depths + j,        b + 2 * CHUNK + lane);
    ASYNC_CP_B32(color + 3 * j + 0, b + 3 * CHUNK + lane);
    ASYNC_CP_B32(color + 3 * j + 1, b + 4 * CHUNK + lane);
    ASYNC_CP_B32(color + 3 * j + 2, b + 5 * CHUNK + lane);
  };

  float carry = 0.f;                       // running tau sum along the ray
  float aR = 0.f, aG = 0.f, aB = 0.f, aD = 0.f, aW = 0.f;

  if (nchunks > 0) issue(s, 0);            // prologue: prefetch chunk 0

  for (int k = 0; k < nchunks; ++k) {
    const int base = s + (k << 5);
    const bool hasnext = (k + 1) < nchunks;        // wave-uniform
    if (hasnext) {
      issue(base + CHUNK, (k + 1) & 1);            // prefetch next chunk
      WAIT_ASYNC(FIELDS);                          // older 6 done, 6 in flight
    } else {
      WAIT_ASYNC(0);
    }

    const float* b = &smem[wave][k & 1][0];
    const int i = base + lane;
    const bool valid = i < e;

    float tau = valid ? b[0 * CHUNK + lane] * b[1 * CHUNK + lane] : 0.f;
    const float dep = b[2 * CHUNK + lane];
    const float cR = b[3 * CHUNK + lane];
    const float cG = b[4 * CHUNK + lane];
    const float cB = b[5 * CHUNK + lane];

    // wave32 inclusive scan of tau (Hillis-Steele over ds_bpermute)
    float incl = tau;
#pragma unroll
    for (int off = 1; off < 32; off <<= 1) {
      float t = bperm_f32(lane - off, incl);
      if (lane >= off) incl += t;
    }
    const float excl = carry + incl - tau;
    // w = exp(-excl) * (1 - exp(-tau)) = exp(-excl) - exp(-(excl+tau))
    float w = __expf(-excl) - __expf(-(excl + tau));
    if (!valid) w = 0.f;

    aR += w * cR; aG += w * cG; aB += w * cB;
    aD += w * dep; aW += w;

    carry += bperm_f32(31, incl);          // chunk total, broadcast
  }

  // wave32 butterfly reduction of the 5 accumulators
#pragma unroll
  for (int off = 16; off > 0; off >>= 1) {
    aR += bperm_f32(lane ^ off, aR);
    aG += bperm_f32(lane ^ off, aG);
    aB += bperm_f32(lane ^ off, aB);
    aD += bperm_f32(lane ^ off, aD);
    aW += bperm_f32(lane ^ off, aW);
  }

  if (lane == 0) {
    const float alpha = aW;
    // white background: rgb = (1 - alpha) + alpha * sum(w*color)
    out[3 * ray + 0] = (1.f - alpha) + alpha * aR;
    out[3 * ray + 1] = (1.f - alpha) + alpha * aG;
    out[3 * ray + 2] = (1.f - alpha) + alpha * aB;
    out[3 * R + ray] = aD;                        // depth
    out[4 * R + ray] = alpha;                     // alpha
    out[5 * R + ray] = (alpha > 0.f) ? 1.f : 0.f; // hit
  }
}

// ---------------------------------------------------------------------------
extern "C" void kernel_launch(void* const* d_in, const int* in_sizes, int n_in,
                              void* d_out, int out_size, void* d_ws, size_t ws_size,
                              hipStream_t stream) {
  const float* color   = (const float*)d_in[0];
  const float* density = (const float*)d_in[1];
  const float* deltas  = (const float*)d_in[2];
  const float* depths  = (const float*)d_in[3];
  const int*   ridx    = (const int*)d_in[4];
  (void)n_in; (void)ws_size;

  const int N = in_sizes[1];          // density has N elements
  const int R = out_size / 6;         // rgb(3R)+depth(R)+alpha(R)+hit(R)

  int* seg_start = (int*)d_ws;        // [R]
  int* seg_end   = seg_start + R;     // [R]

  float* out = (float*)d_out;

  init_seg_kernel<<<(R + 255) / 256, 256, 0, stream>>>(seg_start, seg_end, R);
  mark_bounds_kernel<<<(N + 255) / 256, 256, 0, stream>>>(ridx, N, seg_start, seg_end);
  trace_kernel<<<(R + WAVES_PER_BLOCK - 1) / WAVES_PER_BLOCK, BLOCK, 0, stream>>>(
      color, density, deltas, depths, seg_start, seg_end, out, R);
}